// LengthRegulator_15272903704738
// MI455X (gfx1250) — compile-verified
//
#include <hip/hip_runtime.h>
#include <stdint.h>

// LengthRegulator for MI455X (gfx1250).
// Memory-bound: 384 MiB streamed out, x (48 MiB) stays L2-resident.
// CDNA5 paths used: global_load_async_to_lds_b128 + s_wait_asynccnt (ASYNCcnt),
// LDS scan, b128 gathers, non-temporal b128 stores.

#define B_      32
#define T_      1024
#define D_      384
#define MAXLEN_ 8192
#define FPB     64                 // frames per block
#define NTHR    256                // threads per block (8 wave32)
#define ROWV    (D_ / 4)           // 96 float4 per row

typedef float v4f __attribute__((ext_vector_type(4)));

__global__ __launch_bounds__(NTHR)
void length_regulator_kernel(const float* __restrict__ x,
                             const int* __restrict__ dur,
                             float* __restrict__ out,
                             int* __restrict__ mel_len)
{
    __shared__ __align__(16) int s_csum[T_];   // duration row, scanned in place
    __shared__ int s_part[NTHR];               // per-thread partial sums
    __shared__ int s_idx[FPB];                 // token index per frame (-1 = invalid)

    const int t  = threadIdx.x;
    const int b  = blockIdx.y;
    const int f0 = blockIdx.x * FPB;

    // ---- async-copy duration[b][0..1023] (4 KB) into LDS: 16 B per lane ----
    {
        const int* gbase = dur + (size_t)b * T_;                 // uniform -> SGPR pair
        unsigned   voff  = (unsigned)t * 16u;                    // byte offset per lane
        unsigned   ldsa  = (unsigned)(uintptr_t)&s_csum[t * 4];  // LDS byte address
        asm volatile("global_load_async_to_lds_b128 %0, %1, %2"
                     :
                     : "v"(ldsa), "v"(voff), "s"(gbase)
                     : "memory");
        asm volatile("s_wait_asynccnt 0x0" ::: "memory");
    }
    __syncthreads();

    // ---- in-block inclusive scan of 1024 durations (4/thread + 256-wide scan) ----
    int x0 = s_csum[t * 4 + 0];
    int x1 = s_csum[t * 4 + 1];
    int x2 = s_csum[t * 4 + 2];
    int x3 = s_csum[t * 4 + 3];
    x1 += x0; x2 += x1; x3 += x2;              // local inclusive
    s_part[t] = x3;
    __syncthreads();
    #pragma unroll
    for (int off = 1; off < NTHR; off <<= 1) {
        int add = (t >= off) ? s_part[t - off] : 0;
        __syncthreads();
        s_part[t] += add;
        __syncthreads();
    }
    const int prefix = s_part[t] - x3;         // exclusive prefix of this chunk
    s_csum[t * 4 + 0] = x0 + prefix;
    s_csum[t * 4 + 1] = x1 + prefix;
    s_csum[t * 4 + 2] = x2 + prefix;
    s_csum[t * 4 + 3] = x3 + prefix;
    __syncthreads();

    const int total = s_csum[T_ - 1];          // mel_len for this batch
    if (blockIdx.x == 0 && t == 0)
        mel_len[b] = total;                    // raw i32 bits into d_out tail

    // ---- searchsorted(csum, f, side='right') for the block's 64 frames ----
    if (t < FPB) {
        const int f = f0 + t;
        int lo = 0, hi = T_;
        while (lo < hi) {
            int mid = (lo + hi) >> 1;
            if (s_csum[mid] <= f) lo = mid + 1; else hi = mid;
        }
        int idx = (lo >= T_) ? (T_ - 1) : lo;  // clip to [0, T-1]
        s_idx[t] = (f < total) ? idx : -1;     // -1 => write zeros
    }
    __syncthreads();

    // ---- stream 64 rows x 96 float4 (nontemporal b128 stores) ----
    const v4f* __restrict__ x4 = (const v4f*)x;
    v4f*       __restrict__ o4 = (v4f*)out;
    const size_t xb0   = (size_t)b * T_ * ROWV;
    const size_t orow0 = ((size_t)b * MAXLEN_ + (size_t)f0) * ROWV;

    #pragma unroll 4
    for (int e = t; e < FPB * ROWV; e += NTHR) {
        const int fl  = e / ROWV;              // frame within block
        const int j   = e - fl * ROWV;         // float4 column
        const int idx = s_idx[fl];
        v4f v = (v4f)(0.0f);
        if (idx >= 0)
            v = x4[xb0 + (size_t)idx * ROWV + j];
        __builtin_nontemporal_store(v, &o4[orow0 + (size_t)fl * ROWV + j]);
    }
}

extern "C" void kernel_launch(void* const* d_in, const int* in_sizes, int n_in,
                              void* d_out, int out_size, void* d_ws, size_t ws_size,
                              hipStream_t stream)
{
    (void)in_sizes; (void)n_in; (void)d_ws; (void)ws_size; (void)out_size;

    const float* x   = (const float*)d_in[0];
    const int*   dur = (const int*)d_in[1];
    // d_in[2] = max_len scalar (compile-time constant 8192 here)

    float* out     = (float*)d_out;
    int*   mel_len = (int*)(out + (size_t)B_ * MAXLEN_ * D_);

    dim3 grid(MAXLEN_ / FPB, B_);
    dim3 block(NTHR);
    length_regulator_kernel<<<grid, block, 0, stream>>>(x, dur, out, mel_len);
}